// SparseMatrixModule_34222299415218
// MI455X (gfx1250) — compile-verified
//
#include <hip/hip_runtime.h>

// CSR SpMV tuned for MI455X (gfx1250, wave32):
//   y[r] = sum_j values[j] * x[col[j]],  uniform 32 nnz per row.
// Memory-bound: ~260MB streamed / 23.3 TB/s ~= 11us floor. x (4MB) stays
// resident in the 192MB L2; values/cols are streamed once.
//
// Stages the value/col-index streams through LDS with the CDNA5 async path
// (GLOBAL_LOAD_ASYNC_TO_LDS_B128 + s_wait_asynccnt), double-buffered per
// wave, so each wave has the next chunk's streams in flight while the
// current chunk's x-gathers are outstanding.

typedef __attribute__((ext_vector_type(4))) float v4f;
typedef __attribute__((ext_vector_type(4))) int   v4i;

#if __has_builtin(__builtin_amdgcn_global_load_async_to_lds_b128) && \
    __has_builtin(__builtin_amdgcn_s_wait_asynccnt)
#define HAVE_ASYNC_LDS 1
#else
#define HAVE_ASYNC_LDS 0
#endif

// Exact pointee type of the async builtins' pointer params (clang proto
// "vV4i*1V4i*3IiIi"): GCC-style vector of 4 ints, AS1 (global) / AS3 (LDS).
typedef int b128_t __attribute__((__vector_size__(16)));
typedef __attribute__((address_space(1))) b128_t* async_gp_t;
typedef __attribute__((address_space(3))) b128_t* async_lp_t;

// ---------------------------------------------------------------------------
// Shared math: lane k of a wave holds nnz [4k, 4k+4) of a 128-nnz chunk
// (= 4 rows). nnz 4k belongs to row k/8, so an 8-lane xor-reduce (masks
// 1,2,4) leaves the 4 row sums in lanes 0,8,16,24 -> one coalesced store.
// ---------------------------------------------------------------------------
__device__ __forceinline__ void spmv_chunk_tail(const float* __restrict__ x,
                                                v4f v, v4i c, long chunk,
                                                int lane,
                                                float* __restrict__ y) {
  float p = v.x * x[c.x];
  p = fmaf(v.y, x[c.y], p);
  p = fmaf(v.z, x[c.z], p);
  p = fmaf(v.w, x[c.w], p);
  p += __shfl_xor(p, 1);
  p += __shfl_xor(p, 2);
  p += __shfl_xor(p, 4);
  if ((lane & 7) == 0) {
    y[chunk * 4 + (lane >> 3)] = p;
  }
}

#if HAVE_ASYNC_LDS
// ---------------------------------------------------------------------------
// Async pipeline: per wave, per chunk, each of the 32 lanes DMAs 16B of
// values and 16B of col indices into LDS (2 x GLOBAL_LOAD_ASYNC_TO_LDS_B128
// = 1KB per chunk). Double buffered: issue chunk i+1, then s_wait_asynccnt 2
// retires chunk i's pair (async loads complete in order). Each lane reads
// back exactly the bytes its own async op wrote -> no barrier needed; WAR on
// buffer reuse is ordered by the ds_load waits the FMA consumption forces.
// ---------------------------------------------------------------------------
__global__ __launch_bounds__(256) void spmv_uniform32_async_kernel(
    const float* __restrict__ x,
    const v4f*  __restrict__ vals4,
    const v4i*  __restrict__ cols4,
    float*      __restrict__ y,
    long nChunks)   // nChunks = rows / 4
{
  // 8 waves/block * 2 buffers * (512B vals + 512B cols) = 16KB
  __shared__ __align__(16) unsigned char smem[8 * 2 * 1024];

  const int  lane        = threadIdx.x & 31;
  const int  waveInBlock = threadIdx.x >> 5;
  const long waveId      = (long)blockIdx.x * (blockDim.x >> 5) + waveInBlock;
  const long waveStride  = (long)gridDim.x * (blockDim.x >> 5);

  unsigned char* myBase = &smem[waveInBlock * 2048];

  auto issue = [&](long chunk, int buf) {
    const long base = chunk * 32 + lane;   // float4 / int4 element index
    unsigned char* vdst = myBase + buf * 1024 + lane * 16;
    unsigned char* cdst = myBase + buf * 1024 + 512 + lane * 16;
    __builtin_amdgcn_global_load_async_to_lds_b128(
        (async_gp_t)(vals4 + base), (async_lp_t)vdst, 0, 0);
    __builtin_amdgcn_global_load_async_to_lds_b128(
        (async_gp_t)(cols4 + base), (async_lp_t)cdst, 0, 0);
  };

  long chunk = waveId;
  if (chunk < nChunks) issue(chunk, 0);

  int buf = 0;
  for (; chunk < nChunks; chunk += waveStride) {
    const long next = chunk + waveStride;
    if (next < nChunks) {
      issue(next, buf ^ 1);                  // keep next chunk in flight
      __builtin_amdgcn_s_wait_asynccnt(2);   // current chunk's pair done
    } else {
      __builtin_amdgcn_s_wait_asynccnt(0);
    }

    const v4f v = *(const v4f*)(myBase + buf * 1024 + lane * 16);
    const v4i c = *(const v4i*)(myBase + buf * 1024 + 512 + lane * 16);
    spmv_chunk_tail(x, v, c, chunk, lane, y);
    buf ^= 1;
  }
}
#endif  // HAVE_ASYNC_LDS

// ---------------------------------------------------------------------------
// Direct path (fallback if async builtins unavailable): NT b128 stream loads
// + global_prefetch_b8 one grid-stride ahead.
// ---------------------------------------------------------------------------
__global__ __launch_bounds__(256) void spmv_uniform32_kernel(
    const float* __restrict__ x,
    const v4f*  __restrict__ vals4,
    const v4i*  __restrict__ cols4,
    float*      __restrict__ y,
    long nChunks)
{
  const int  lane        = threadIdx.x & 31;
  const int  waveInBlock = threadIdx.x >> 5;
  const long waveId      = (long)blockIdx.x * (blockDim.x >> 5) + waveInBlock;
  const long waveStride  = (long)gridDim.x * (blockDim.x >> 5);
  const long lastIdx     = nChunks * 32;

  for (long chunk = waveId; chunk < nChunks; chunk += waveStride) {
    const long base = chunk * 32 + lane;

    v4f v = __builtin_nontemporal_load(&vals4[base]);
    v4i c = __builtin_nontemporal_load(&cols4[base]);

    const long nbase = base + waveStride * 32;
    if (nbase < lastIdx) {
      __builtin_prefetch((const void*)(vals4 + nbase), 0, 1);
      __builtin_prefetch((const void*)(cols4 + nbase), 0, 1);
    }

    spmv_chunk_tail(x, v, c, chunk, lane, y);
  }
}

// ---------------------------------------------------------------------------
// Generic fallback: one wave per row driven by row_ptrs (any nnz per row).
// ---------------------------------------------------------------------------
__global__ __launch_bounds__(256) void spmv_generic_kernel(
    const float* __restrict__ x,
    const float* __restrict__ vals,
    const int*   __restrict__ cols,
    const int*   __restrict__ row_ptrs,
    float*       __restrict__ y,
    long rows)
{
  const int  lane       = threadIdx.x & 31;
  const long wave       = ((long)blockIdx.x * blockDim.x + threadIdx.x) >> 5;
  const long waveStride = ((long)gridDim.x * blockDim.x) >> 5;

  for (long r = wave; r < rows; r += waveStride) {
    const int s = row_ptrs[r];
    const int e = row_ptrs[r + 1];
    float acc = 0.0f;
    for (int j = s + lane; j < e; j += 32) {
      acc = fmaf(vals[j], x[cols[j]], acc);
    }
    acc += __shfl_xor(acc, 16);
    acc += __shfl_xor(acc, 8);
    acc += __shfl_xor(acc, 4);
    acc += __shfl_xor(acc, 2);
    acc += __shfl_xor(acc, 1);
    if (lane == 0) y[r] = acc;
  }
}

extern "C" void kernel_launch(void* const* d_in, const int* in_sizes, int n_in,
                              void* d_out, int out_size, void* d_ws, size_t ws_size,
                              hipStream_t stream) {
  (void)n_in; (void)d_ws; (void)ws_size; (void)out_size;

  const float* x        = (const float*)d_in[0];
  const float* vals     = (const float*)d_in[1];
  const int*   cols     = (const int*)d_in[2];
  const int*   row_ptrs = (const int*)d_in[3];
  float*       y        = (float*)d_out;

  const long nnz  = (long)in_sizes[1];
  const long rows = (long)in_sizes[3] - 1;

  const int block = 256;           // 8 waves per block (wave32)

  if (rows > 0 && (rows & 3) == 0 && nnz == rows * 32) {
    const long nChunks       = rows >> 2;            // 250,000 for 1M rows
    const long wavesPerBlock = block >> 5;           // 8
    long grid = (nChunks + wavesPerBlock - 1) / wavesPerBlock;
    if (grid > 8192) grid = 8192;                    // grid-stride pipeline
#if HAVE_ASYNC_LDS
    spmv_uniform32_async_kernel<<<(int)grid, block, 0, stream>>>(
        x, (const v4f*)vals, (const v4i*)cols, y, nChunks);
#else
    spmv_uniform32_kernel<<<(int)grid, block, 0, stream>>>(
        x, (const v4f*)vals, (const v4i*)cols, y, nChunks);
#endif
  } else {
    long grid = (rows * 32 + block - 1) / block;     // one wave per row
    if (grid > 16384) grid = 16384;
    if (grid < 1) grid = 1;
    spmv_generic_kernel<<<(int)grid, block, 0, stream>>>(
        x, vals, cols, row_ptrs, y, rows);
  }
}